// EMDLoss_17884243821447
// MI455X (gfx1250) — compile-verified
//
#include <hip/hip_runtime.h>
#include <stdint.h>

// ---------------- problem constants ----------------
#define NUM_BINS 256
#define NBATCH   16
#define CHW      (3 * 512 * 512)       // 786432 floats per batch item
#define N4       (CHW / 4)             // 196608 float4 per batch item
#define BLOCKS_X 64
#define THREADS  256
#define WAVES    (THREADS / 32)        // 8 waves (wave32)
#define STRIDE   (BLOCKS_X * THREADS)  // 16384 float4 per grid-x sweep
#define SWEEPS   (N4 / STRIDE)         // 12 sweeps total, exact
#define ITERS    (SWEEPS / 2)          // 6 pipeline iterations, 2 float4/lane each
#define NSTAGE   4                     // async ring depth (3 stages in flight)
#define NCOPY    16                    // half-wave-private histogram copies

typedef __attribute__((address_space(3))) void* lds_vp;

// Async copy 16 bytes/lane from global to this lane's LDS slot.
// GVS mode: saddr(64-bit SGPR base) + vaddr(32-bit per-lane byte offset).
__device__ __forceinline__ void async_ld_b128(void* lds_dst, uint32_t voff_bytes,
                                              const float* sbase) {
    asm volatile("global_load_async_to_lds_b128 %0, %1, %2"
                 :
                 : "v"((lds_vp)lds_dst), "v"(voff_bytes), "s"(sbase)
                 : "memory");
}

// Wait until the oldest in-flight stage (2 async ops) has landed.
// rem = stages issued after the one we are about to consume (compile-time).
__device__ __forceinline__ void wait_async_stage(int rem) {
    if (rem >= 3)      asm volatile("s_wait_asynccnt 6" ::: "memory");
    else if (rem == 2) asm volatile("s_wait_asynccnt 4" ::: "memory");
    else if (rem == 1) asm volatile("s_wait_asynccnt 2" ::: "memory");
    else               asm volatile("s_wait_asynccnt 0" ::: "memory");
}

// ---------------- kernel 0: zero the global histograms ----------------
__global__ void zero_hist_kernel(uint32_t* __restrict__ g) {
    g[blockIdx.x * blockDim.x + threadIdx.x] = 0u;
}

// ---------------- kernel 1: per-batch histograms ----------------
// grid = (BLOCKS_X, NBATCH, 2 images); block = 256 threads = 8 waves.
// ghist layout: [img][batch][bin] as u32 counts.
__global__ __launch_bounds__(THREADS)
void hist_kernel(const float* __restrict__ im1, const float* __restrict__ im2,
                 uint32_t* __restrict__ ghist) {
    __shared__ uint32_t hist[NCOPY * NUM_BINS];      // 16 KB: half-wave-private hists
    __shared__ float4   stage[WAVES][NSTAGE][2][32]; // 32 KB: 4-deep async ring

    const int tid   = threadIdx.x;
    const int w     = tid >> 5;       // wave id within block
    const int lane  = tid & 31;
    const int batch = blockIdx.y;
    const int img   = blockIdx.z;

    // zero the histogram copies (4096 u32 across 256 threads)
#pragma unroll
    for (int k = 0; k < NCOPY; ++k) hist[k * NUM_BINS + tid] = 0u;
    __syncthreads();

    const float* src  = (img == 0) ? im1 : im2;
    const float* base = src + (size_t)batch * CHW;              // uniform -> SGPRs
    const uint32_t i0 = (uint32_t)(blockIdx.x * THREADS + tid); // lane's float4 index

    uint32_t* myhist = &hist[(tid >> 4) * NUM_BINS];  // private per 16 lanes

    // prologue: fill 3 of 4 ring slots (2 async B128 per lane per stage)
#pragma unroll
    for (int s = 0; s < NSTAGE - 1; ++s) {
        async_ld_b128(&stage[w][s][0][lane], (i0 + (uint32_t)s * STRIDE) * 16u, base);
        async_ld_b128(&stage[w][s][1][lane],
                      (i0 + (uint32_t)(s + ITERS) * STRIDE) * 16u, base);
    }

#pragma unroll
    for (int it = 0; it < ITERS; ++it) {
        if (it + NSTAGE - 1 < ITERS) {
            const int s = (it + NSTAGE - 1) & (NSTAGE - 1);
            async_ld_b128(&stage[w][s][0][lane],
                          (i0 + (uint32_t)(it + NSTAGE - 1) * STRIDE) * 16u, base);
            async_ld_b128(&stage[w][s][1][lane],
                          (i0 + (uint32_t)(it + NSTAGE - 1 + ITERS) * STRIDE) * 16u, base);
        }
        wait_async_stage(ITERS - 1 - it);

        const int cs = it & (NSTAGE - 1);
        float4 va = stage[w][cs][0][lane];   // ds_load_b128
        float4 vb = stage[w][cs][1][lane];   // ds_load_b128

        // exact reference binning: clip(floor(x*255 * (256/255)), 0, 255)
        float t; int b;
        t = va.x * 255.0f; b = (int)floorf(t * (256.0f / 255.0f));
        b = b < 0 ? 0 : (b > 255 ? 255 : b); atomicAdd(&myhist[b], 1u);
        t = va.y * 255.0f; b = (int)floorf(t * (256.0f / 255.0f));
        b = b < 0 ? 0 : (b > 255 ? 255 : b); atomicAdd(&myhist[b], 1u);
        t = va.z * 255.0f; b = (int)floorf(t * (256.0f / 255.0f));
        b = b < 0 ? 0 : (b > 255 ? 255 : b); atomicAdd(&myhist[b], 1u);
        t = va.w * 255.0f; b = (int)floorf(t * (256.0f / 255.0f));
        b = b < 0 ? 0 : (b > 255 ? 255 : b); atomicAdd(&myhist[b], 1u);
        t = vb.x * 255.0f; b = (int)floorf(t * (256.0f / 255.0f));
        b = b < 0 ? 0 : (b > 255 ? 255 : b); atomicAdd(&myhist[b], 1u);
        t = vb.y * 255.0f; b = (int)floorf(t * (256.0f / 255.0f));
        b = b < 0 ? 0 : (b > 255 ? 255 : b); atomicAdd(&myhist[b], 1u);
        t = vb.z * 255.0f; b = (int)floorf(t * (256.0f / 255.0f));
        b = b < 0 ? 0 : (b > 255 ? 255 : b); atomicAdd(&myhist[b], 1u);
        t = vb.w * 255.0f; b = (int)floorf(t * (256.0f / 255.0f));
        b = b < 0 ? 0 : (b > 255 ? 255 : b); atomicAdd(&myhist[b], 1u);
    }

    __syncthreads();
    // fold the 16 private copies; one global atomic per bin per block
    uint32_t s = 0;
#pragma unroll
    for (int k = 0; k < NCOPY; ++k) s += hist[k * NUM_BINS + tid];
    atomicAdd(&ghist[(((img * NBATCH) + batch) << 8) + tid], s);
}

// ---------------- kernel 2: cumsum + EMD + final scalar ----------------
// single wave32; lanes 0..15 each own one batch item.
__global__ void finish_kernel(const uint32_t* __restrict__ ghist,
                              float* __restrict__ out) {
    const int lane = threadIdx.x;
    float emd = 0.0f;
    if (lane < NBATCH) {
        const uint32_t* h1 = ghist + lane * NUM_BINS;
        const uint32_t* h2 = ghist + (NBATCH + lane) * NUM_BINS;
        int c1 = 0, c2 = 0;
        for (int bin = 0; bin < NUM_BINS; ++bin) {
            c1 += (int)h1[bin];
            c2 += (int)h2[bin];
            emd += fabsf((float)(c1 - c2));   // exact integer cum-diff
        }
        emd *= (1.0f / (float)CHW);           // histogram normalization
    }
#pragma unroll
    for (int off = 8; off > 0; off >>= 1) emd += __shfl_down(emd, off, 16);
    if (lane == 0) out[0] = emd / (float)NBATCH / 3.0f;   // SCALAR == 1
}

// ---------------- launcher ----------------
extern "C" void kernel_launch(void* const* d_in, const int* in_sizes, int n_in,
                              void* d_out, int out_size, void* d_ws, size_t ws_size,
                              hipStream_t stream) {
    const float* im1 = (const float*)d_in[0];
    const float* im2 = (const float*)d_in[1];
    float* out       = (float*)d_out;
    uint32_t* ghist  = (uint32_t*)d_ws;   // 2*16*256 u32 = 32 KB scratch

    zero_hist_kernel<<<(2 * NBATCH * NUM_BINS) / THREADS, THREADS, 0, stream>>>(ghist);

    dim3 grid(BLOCKS_X, NBATCH, 2);
    hist_kernel<<<grid, THREADS, 0, stream>>>(im1, im2, ghist);

    finish_kernel<<<1, 32, 0, stream>>>(ghist, out);
}